// DOSAConLoss_54855322304796
// MI455X (gfx1250) — compile-verified
//
#include <hip/hip_runtime.h>

typedef __attribute__((ext_vector_type(2))) float v2f;
typedef __attribute__((ext_vector_type(8))) float v8f;

#define NBINS 1024
#define NBLOCKS 1024
#define BLKTHREADS 256
#define CIOU_EPS 1e-7f
#define FOUR_OVER_PI2 0.40528473456935108f

__device__ __forceinline__ float ciou_base(float4 p, float4 t) {
    // xywh boxes
    float x1 = p.x, y1 = p.y, w1 = p.z, h1 = p.w;
    float x2 = t.x, y2 = t.y, w2 = t.z, h2 = t.w;
    float b1x1 = x1 - w1 * 0.5f, b1x2 = x1 + w1 * 0.5f;
    float b1y1 = y1 - h1 * 0.5f, b1y2 = y1 + h1 * 0.5f;
    float b2x1 = x2 - w2 * 0.5f, b2x2 = x2 + w2 * 0.5f;
    float b2y1 = y2 - h2 * 0.5f, b2y2 = y2 + h2 * 0.5f;

    float iw = fmaxf(fminf(b1x2, b2x2) - fmaxf(b1x1, b2x1), 0.0f);
    float ih = fmaxf(fminf(b1y2, b2y2) - fmaxf(b1y1, b2y1), 0.0f);
    float inter = iw * ih;
    float uni   = w1 * h1 + w2 * h2 - inter + CIOU_EPS;
    float iou   = inter / uni;

    float cw = fmaxf(b1x2, b2x2) - fminf(b1x1, b2x1);
    float ch = fmaxf(b1y2, b2y2) - fminf(b1y1, b2y1);
    float c2 = cw * cw + ch * ch + CIOU_EPS;

    // ((b2x1+b2x2-b1x1-b1x2)^2 + (...)^2) * 0.25 == (x2-x1)^2 + (y2-y1)^2
    float dx = x2 - x1, dy = y2 - y1;
    float rho2 = dx * dx + dy * dy;

    float da = atanf(w2 / h2) - atanf(w1 / h1);
    float v  = FOUR_OVER_PI2 * da * da;
    float a  = v / (v - iou + (1.0f + CIOU_EPS));   // stop_gradient: fwd identical

    float ciou  = iou - (rho2 / c2 + v * a);
    float one_m = 1.0f - ciou;
    float sw    = 1.0f / (w2 * h2 + 1e-7f);
    return one_m * one_m * one_m * sw;              // (1-iou)^GAMMA * scale_weight
}

__device__ __forceinline__ void hist_add(unsigned int* shist, float cx, float cy) {
    int gx = min(max((int)(cx * 32.0f), 0), 31);
    int gy = min(max((int)(cy * 32.0f), 0), 31);
    atomicAdd(&shist[gy * 32 + gx], 1u);
}

__global__ __launch_bounds__(NBINS) void dosa_hist_init(unsigned int* __restrict__ ghist) {
    ghist[threadIdx.x] = 0u;
}

__global__ __launch_bounds__(BLKTHREADS) void dosa_main(
    const float4* __restrict__ pred, const float4* __restrict__ targ,
    long long nBoxes, float* __restrict__ partials, unsigned int* __restrict__ ghist)
{
    __shared__ unsigned int shist[NBINS];
    __shared__ float wsum[BLKTHREADS / 32];

    for (int i = threadIdx.x; i < NBINS; i += blockDim.x) shist[i] = 0u;
    __syncthreads();

    const long long total = (long long)gridDim.x * blockDim.x;   // threads in grid
    const long long tid   = (long long)blockIdx.x * blockDim.x + threadIdx.x;
    const long long chunk = total << 1;                          // boxes per full pass
    const long long fullIters = nBoxes / chunk;                  // uniform, branch-free body
    const long long rem = nBoxes - fullIters * chunk;            // tail boxes (uniform)

    v8f acc = {0.f, 0.f, 0.f, 0.f, 0.f, 0.f, 0.f, 0.f};
    const v2f ones = {1.0f, 1.0f};

    const float4* __restrict__ pp = pred + (tid << 1);
    const float4* __restrict__ tp = targ + (tid << 1);

    // ---- main loop: no validity masks, no exec churn ----
    for (long long it = 0; it < fullIters; ++it) {
        __builtin_prefetch((const void*)(pp + chunk), 0, 3);
        __builtin_prefetch((const void*)(tp + chunk), 0, 3);

        float4 p0 = pp[0], p1 = pp[1];
        float4 t0 = tp[0], t1 = tp[1];

        float b0 = ciou_base(p0, t0);
        float b1 = ciou_base(p1, t1);

        hist_add(shist, t0.x, t0.y);
        hist_add(shist, t1.x, t1.y);

        // EXEC all-1s (no divergence in this loop).
        // D = A*ones + C: 64 fresh fp32 values folded into 256 fp32
        // accumulators per call on the matrix pipe; sum(D) = 16*sum(A-fed).
        v2f a;
        a.x = b0;
        a.y = b1;
        acc = __builtin_amdgcn_wmma_f32_16x16x4_f32(
            false, a, false, ones, (short)0, acc, false, false);

        pp += chunk;
        tp += chunk;
    }

    // ---- single guarded tail pass (uniform predicate; EXEC full at WMMA) ----
    if (rem > 0) {
        long long i0 = fullIters * chunk + (tid << 1);
        long long i1 = i0 + 1;
        bool v0 = (i0 < nBoxes);
        bool v1 = (i1 < nBoxes);
        long long c0 = v0 ? i0 : 0;
        long long c1 = v1 ? i1 : 0;

        float4 p0 = pred[c0], t0 = targ[c0];
        float4 p1 = pred[c1], t1 = targ[c1];

        float b0 = v0 ? ciou_base(p0, t0) : 0.0f;
        float b1 = v1 ? ciou_base(p1, t1) : 0.0f;

        if (v0) hist_add(shist, t0.x, t0.y);
        if (v1) hist_add(shist, t1.x, t1.y);

        v2f a;
        a.x = b0;
        a.y = b1;
        acc = __builtin_amdgcn_wmma_f32_16x16x4_f32(
            false, a, false, ones, (short)0, acc, false, false);
    }

    // ---- wave + block reduction of the WMMA accumulators ----
    float s = acc[0] + acc[1] + acc[2] + acc[3] + acc[4] + acc[5] + acc[6] + acc[7];
    #pragma unroll
    for (int off = 16; off > 0; off >>= 1) s += __shfl_down(s, off, 32);

    int wid  = threadIdx.x >> 5;
    int lane = threadIdx.x & 31;
    if (lane == 0) wsum[wid] = s;
    __syncthreads();

    if (threadIdx.x == 0) {
        float bs = 0.0f;
        #pragma unroll
        for (int w = 0; w < BLKTHREADS / 32; ++w) bs += wsum[w];
        partials[blockIdx.x] = bs * (1.0f / 16.0f);   // undo 16-column replication
    }
    __syncthreads();

    // flush privatized histogram to global
    for (int i = threadIdx.x; i < NBINS; i += blockDim.x) {
        unsigned int c = shist[i];
        if (c) atomicAdd(&ghist[i], c);
    }
}

__global__ __launch_bounds__(NBINS) void dosa_final(
    const float* __restrict__ partials, const unsigned int* __restrict__ ghist,
    long long nBoxes, float* __restrict__ out)
{
    __shared__ float ssum[NBINS];
    __shared__ unsigned int smax[NBINS];
    int t = threadIdx.x;
    ssum[t] = partials[t];      // NBLOCKS == NBINS == 1024
    smax[t] = ghist[t];
    __syncthreads();
    #pragma unroll
    for (int off = NBINS / 2; off > 0; off >>= 1) {
        if (t < off) {
            ssum[t] += ssum[t + off];
            smax[t] = max(smax[t], smax[t + off]);
        }
        __syncthreads();
    }
    if (t == 0) {
        float mean_base = ssum[0] / (float)nBoxes;
        float maxh = (float)smax[0];
        // mean(1 + ALPHA*hist/max) = 1 + ALPHA*(N/1024)/max  (sum(hist)==N)
        float mean_dw = 1.0f + 1.5f * ((float)nBoxes / 1024.0f) / maxh;
        out[0] = mean_base * mean_dw;
    }
}

extern "C" void kernel_launch(void* const* d_in, const int* in_sizes, int n_in,
                              void* d_out, int out_size, void* d_ws, size_t ws_size,
                              hipStream_t stream) {
    const float4* pred = (const float4*)d_in[0];
    const float4* targ = (const float4*)d_in[1];
    long long nBoxes = (long long)in_sizes[0] / 4;   // (N,4) float32

    unsigned int* ghist = (unsigned int*)d_ws;
    float* partials = (float*)((char*)d_ws + NBINS * sizeof(unsigned int));

    dosa_hist_init<<<1, NBINS, 0, stream>>>(ghist);
    dosa_main<<<NBLOCKS, BLKTHREADS, 0, stream>>>(pred, targ, nBoxes, partials, ghist);
    dosa_final<<<1, NBINS, 0, stream>>>(partials, ghist, nBoxes, (float*)d_out);
}